// GradientForceOutput_8821862826155
// MI455X (gfx1250) — compile-verified
//
#include <hip/hip_runtime.h>
#include <stdint.h>

#define TILE 256
#define NBLOCKS 3072

typedef float f32x3 __attribute__((ext_vector_type(3))) __attribute__((aligned(4)));

// E = sum_e a_e*r2 + b_e*exp(-r2);  force_e = -dE/ddisp = 2*disp*(b*exp(-r2) - a)
// out[src] += f ; out[dst] -= f   (fuses f_src - f_dst into one buffer)
__global__ __launch_bounds__(256) void edge_scatter_force(
    const float* __restrict__ disp,
    const float* __restrict__ a,
    const float* __restrict__ b,
    const long long* __restrict__ e_src,
    const long long* __restrict__ e_dst,
    float* __restrict__ out,
    int n_edges)
{
    __shared__ float     sa[2][TILE];
    __shared__ float     sb[2][TILE];
    __shared__ long long ss[2][TILE];
    __shared__ long long sd[2][TILE];

    const int tid = threadIdx.x;
    const long long stride = (long long)gridDim.x * TILE;
    const long long base0  = (long long)blockIdx.x * TILE;

    // Async global->LDS prefetch of one tile of a[], b[], src-idx[], dst-idx[]
    // (GVS mode: sgpr base + 32-bit vgpr byte offset). Each lane copies its own
    // slot, so no cross-lane dependency -> no barrier, only s_wait_asynccnt.
    auto prefetch = [&](long long tbase, int buf) {
        long long idx = tbase + tid;
        if (idx > (long long)n_edges - 1) idx = (long long)n_edges - 1;
        unsigned off4  = (unsigned)idx * 4u;
        unsigned off8  = (unsigned)idx * 8u;
        unsigned lds_a = (unsigned)(unsigned long long)(uintptr_t)&sa[buf][tid];
        unsigned lds_b = (unsigned)(unsigned long long)(uintptr_t)&sb[buf][tid];
        unsigned lds_s = (unsigned)(unsigned long long)(uintptr_t)&ss[buf][tid];
        unsigned lds_d = (unsigned)(unsigned long long)(uintptr_t)&sd[buf][tid];
        asm volatile("global_load_async_to_lds_b32 %0, %1, %2"
                     :: "v"(lds_a), "v"(off4), "s"(a) : "memory");
        asm volatile("global_load_async_to_lds_b32 %0, %1, %2"
                     :: "v"(lds_b), "v"(off4), "s"(b) : "memory");
        asm volatile("global_load_async_to_lds_b64 %0, %1, %2"
                     :: "v"(lds_s), "v"(off8), "s"(e_src) : "memory");
        asm volatile("global_load_async_to_lds_b64 %0, %1, %2"
                     :: "v"(lds_d), "v"(off8), "s"(e_dst) : "memory");
    };

    int buf = 0;
    prefetch(base0, 0);

    for (long long base = base0; base < (long long)n_edges; base += stride) {
        long long nxt = base + stride;               // block-uniform condition
        if (nxt < (long long)n_edges) {
            prefetch(nxt, buf ^ 1);                  // next tile's 4 in flight
            asm volatile("s_wait_asynccnt 0x4" ::: "memory"); // current tile done
        } else {
            asm volatile("s_wait_asynccnt 0x0" ::: "memory");
        }

        long long e = base + tid;
        if (e < (long long)n_edges) {
            float ae = sa[buf][tid];
            float be = sb[buf][tid];
            long long si = ss[buf][tid];
            long long di = sd[buf][tid];

            // single NT b96 load of the 12-byte displacement
            f32x3 d = __builtin_nontemporal_load((const f32x3*)(disp + 3ll * e));
            float dx = d.x, dy = d.y, dz = d.z;

            float r2 = dx * dx + dy * dy + dz * dz;
            float s2 = 2.0f * (be * __expf(-r2) - ae);
            float fx = s2 * dx, fy = s2 * dy, fz = s2 * dz;

            int so   = 3 * (int)si;
            int dofs = 3 * (int)di;
            // Non-returning relaxed agent-scope f32 atomics -> global_atomic_add_f32
            // resolving in L2 (output is 1.2 MB, fully L2-resident).
            __hip_atomic_fetch_add(&out[so + 0],    fx, __ATOMIC_RELAXED, __HIP_MEMORY_SCOPE_AGENT);
            __hip_atomic_fetch_add(&out[so + 1],    fy, __ATOMIC_RELAXED, __HIP_MEMORY_SCOPE_AGENT);
            __hip_atomic_fetch_add(&out[so + 2],    fz, __ATOMIC_RELAXED, __HIP_MEMORY_SCOPE_AGENT);
            __hip_atomic_fetch_add(&out[dofs + 0], -fx, __ATOMIC_RELAXED, __HIP_MEMORY_SCOPE_AGENT);
            __hip_atomic_fetch_add(&out[dofs + 1], -fy, __ATOMIC_RELAXED, __HIP_MEMORY_SCOPE_AGENT);
            __hip_atomic_fetch_add(&out[dofs + 2], -fz, __ATOMIC_RELAXED, __HIP_MEMORY_SCOPE_AGENT);
        }
        buf ^= 1;
    }
}

extern "C" void kernel_launch(void* const* d_in, const int* in_sizes, int n_in,
                              void* d_out, int out_size, void* d_ws, size_t ws_size,
                              hipStream_t stream) {
    // setup_inputs order: disp [E*3] f32, a [E] f32, b [E] f32,
    //                     edge_index [2*E] i64, atom_node [N] i64 (shape only)
    const float*     disp  = (const float*)d_in[0];
    const float*     a     = (const float*)d_in[1];
    const float*     b     = (const float*)d_in[2];
    const long long* eidx  = (const long long*)d_in[3];
    const int n_edges      = in_sizes[1];
    const long long* e_src = eidx;
    const long long* e_dst = eidx + n_edges;
    float* out = (float*)d_out;

    // Output is poisoned by the harness; zero it every call (capture-safe).
    hipMemsetAsync(out, 0, (size_t)out_size * sizeof(float), stream);

    int max_blocks = (n_edges + TILE - 1) / TILE;
    int blocks = NBLOCKS < max_blocks ? NBLOCKS : max_blocks;
    edge_scatter_force<<<blocks, 256, 0, stream>>>(disp, a, b, e_src, e_dst, out, n_edges);
}